// PatchEmbed_61443802137309
// MI455X (gfx1250) — compile-verified
//
#include <hip/hip_runtime.h>
#include <math.h>
#include <stdint.h>

// ---------------------------------------------------------------------------
// CDNA5 (gfx1250) pipeline: embed -> conv1 -> conv2 -> topo/MST branch.
//  * embedding+conv1 folded into per-tap vocab->cout table (8MB, L2 resident)
//  * conv2 / lin2 via V_WMMA_F32_16X16X4_F32 (exact fp32 on the matrix pipe)
//  * conv2 B panel staged by the Tensor Data Mover (TENSOR_LOAD_TO_LDS,
//    TENSORcnt), double buffered, with TDM pad fields producing a
//    bank-conflict-free 260-float LDS row stride. 8 M-tile waves share each
//    panel (8x less L2 traffic) and read it with ds_load_b64.
//  * conv1 token-id staging uses GLOBAL_LOAD_ASYNC_TO_LDS_B32 (ASYNCcnt).
// ---------------------------------------------------------------------------

typedef __attribute__((ext_vector_type(2))) float v2f;
typedef __attribute__((ext_vector_type(4))) float v4f;
typedef __attribute__((ext_vector_type(8))) float v8f;
typedef unsigned int u32x4 __attribute__((ext_vector_type(4)));
typedef int i32x4 __attribute__((ext_vector_type(4)));
typedef int i32x8 __attribute__((ext_vector_type(8)));

__device__ __forceinline__ v8f wmma_f32x4(v2f a, v2f b, v8f c) {
  return __builtin_amdgcn_wmma_f32_16x16x4_f32(false, a, false, b, (short)0, c,
                                               false, false);
}

// Problem constants
#define BATCH 32
#define NTOK 37632
#define L1OUT 2351   // (37632-32)/16+1
#define L2OUT 144    // (2351-63)/16+1
#define CIN 128
#define C1 256
#define C2 512
#define K1 32
#define K2 63
#define VOCAB 256

// Workspace layout (floats)
#define WS_T 0                                        // [32][256][256] table
#define WS_H1 (WS_T + (size_t)K1 * VOCAB * C1)        // [32][2351][256]
#define WS_PTS (WS_H1 + (size_t)BATCH * L1OUT * C1)   // [32][20][3]
#define WS_T1 (WS_PTS + (size_t)BATCH * 20 * 3)       // [32][144]
#define XSEQ_ELEMS ((size_t)BATCH * L2OUT * C2)

// ---------------------------------------------------------------------------
// K1: T[k][v][co] = sum_c emb[v][c] * w1[co][c][k]   (32 GEMMs 256x256x128)
// grid (32,16,16), block 32 — one wave per 16x16 tile.
// ---------------------------------------------------------------------------
__global__ void build_table_kernel(const float* __restrict__ emb,
                                   const float* __restrict__ w1,
                                   float* __restrict__ T) {
  const int k = blockIdx.x;
  const int v0 = blockIdx.y * 16;
  const int co0 = blockIdx.z * 16;
  const int lane = threadIdx.x & 31;
  const int lo = lane & 15;
  const int hf = lane >> 4;

  v8f acc = {};
  #pragma unroll 4
  for (int c0 = 0; c0 < CIN; c0 += 4) {
    const int ca = c0 + 2 * hf, cb = ca + 1;
    v2f a, b;
    a.x = emb[(v0 + lo) * CIN + ca];
    a.y = emb[(v0 + lo) * CIN + cb];
    b.x = w1[((co0 + lo) * CIN + ca) * K1 + k];
    b.y = w1[((co0 + lo) * CIN + cb) * K1 + k];
    acc = wmma_f32x4(a, b, acc);
  }
  #pragma unroll
  for (int j = 0; j < 8; ++j) {
    const int m = j + 8 * hf;
    T[((size_t)k * VOCAB + (v0 + m)) * C1 + co0 + lo] = acc[j];
  }
}

// ---------------------------------------------------------------------------
// K2: H1[b][l][c] = sum_k T[k][ x[b][16l+k] ][c]
// grid (2351, 32), block 64; token ids staged via async global->LDS b32.
// ---------------------------------------------------------------------------
__global__ void conv1_gather_kernel(const int* __restrict__ x,
                                    const float* __restrict__ T,
                                    float* __restrict__ H1) {
  const int l = blockIdx.x;
  const int b = blockIdx.y;
  const int tid = threadIdx.x;          // 0..63
  __shared__ int ids[K1];
  if (tid < K1) {                       // wave 0 issues 32 async id copies
    const char* gb = (const char*)(x + (size_t)b * NTOK + 16 * l);
    asm volatile("global_load_async_to_lds_b32 %0, %1, %2"
                 :: "v"((uint32_t)(uintptr_t)&ids[tid]),
                    "v"((uint32_t)(tid * 4)), "s"(gb)
                 : "memory");
  }
  asm volatile("s_wait_asynccnt 0" ::: "memory");
  __syncthreads();
  v4f acc = {};
  #pragma unroll
  for (int k = 0; k < K1; ++k)
    acc += *(const v4f*)&T[((size_t)k * VOCAB + ids[k]) * C1 + 4 * tid];
  *(v4f*)&H1[((size_t)b * L1OUT + l) * C1 + 4 * tid] = acc;
}

// ---------------------------------------------------------------------------
// K3: conv2 as WMMA GEMM: x_seq[b][l][co] = sum_{k<63,c<256} W2[co][c][k]*H1[b][16l+k][c]
// Block = 256 threads (8 waves), 128 co x 48 l per block. Wave w owns co-tile
// m0+16w for all 48 columns (3 accumulators, A regs reused 3x).
// B panel (48 rows x 256 f32, row stride 16*256 elems) moved global->LDS by
// the TDM: one TENSOR_LOAD_TO_LDS per k-slice (wave 0 issues), double
// buffered; TDM pad fields (interval=256 dwords, amount=4 dwords) give a
// 260-float LDS row stride -> conflict-free ds_load_b64 panel reads.
// grid (3, 4, 32), block 256.
// ---------------------------------------------------------------------------
#define BSTRIDE 260

__device__ __forceinline__ void tdm_load_2d(uint32_t lds_addr, uint64_t gaddr,
                                            i32x8 g1) {
  u32x4 g0;
  g0[0] = 1u;                                        // count=1, user mode
  g0[1] = lds_addr;                                  // LDS byte address
  g0[2] = (uint32_t)gaddr;                           // global addr [31:0]
  g0[3] = (uint32_t)((gaddr >> 32) & 0x01FFFFFFu) |  // global addr [56:32]
          (2u << 30);                                // type = 2 ("image")
  i32x4 z4 = {0, 0, 0, 0};
#if __clang_major__ >= 23
  i32x8 z8 = {0, 0, 0, 0, 0, 0, 0, 0};
  __builtin_amdgcn_tensor_load_to_lds(g0, g1, z4, z4, z8, 0);
#else
  __builtin_amdgcn_tensor_load_to_lds(g0, g1, z4, z4, 0);
#endif
}

__global__ __launch_bounds__(256) void conv2_wmma_kernel(
    const float* __restrict__ H1, const float* __restrict__ W2,
    float* __restrict__ xseq) {
  const int n0 = blockIdx.x * 48;
  const int m0 = blockIdx.y * 128;
  const int b = blockIdx.z;
  const int tid = threadIdx.x;
  const int lane = tid & 31;
  const int wave = tid >> 5;
  const int lo = lane & 15;
  const int hf = lane >> 4;

  __shared__ float bs[2][48 * BSTRIDE];

  // D# group1 (constant across k): data_size=4B; pad_enable, interval=256
  // dwords (one row), amount=4 dwords; tensor_dim0=tile_dim0=256,
  // tensor_dim1=tile_dim1=48 rows; row stride 4096 elements (16*256).
  i32x8 g1;
  g1[0] = (int)((2u << 16) | (1u << 20) | (7u << 22) | (3u << 25));
  g1[1] = (int)(256u << 16);   // tensor_dim0[15:0] in bits[31:16]
  g1[2] = (int)(48u << 16);    // tensor_dim1[15:0] in bits[31:16]
  g1[3] = (int)(256u << 16);   // tile_dim0 in bits[31:16]
  g1[4] = 48;                  // tile_dim1
  g1[5] = 4096;                // tensor_dim0_stride[31:0]
  g1[6] = 0;
  g1[7] = 0;

  // tile start row (k=0): b*2351 + 16*n0 ; +k advances one 1KB row
  const uint64_t gA0 =
      (uint64_t)(uintptr_t)(H1 + ((size_t)b * L1OUT + 16 * (size_t)n0) * C1);

  const float* pa = W2 + (size_t)(m0 + wave * 16 + lo) * (C1 * K2);

  v8f acc0 = {}, acc1 = {}, acc2 = {};

  if (wave == 0)
    tdm_load_2d((uint32_t)(uintptr_t)&bs[0][0], gA0, g1);
  __builtin_amdgcn_s_wait_tensorcnt(0);
  __syncthreads();

  for (int k = 0; k < K2; ++k) {
    if (k + 1 < K2 && wave == 0)   // overlap TDM copy of k+1 with compute on k
      tdm_load_2d((uint32_t)(uintptr_t)&bs[(k + 1) & 1][0],
                  gA0 + (uint64_t)(k + 1) * (C1 * 4), g1);
    const float* bsc = &bs[k & 1][0];
    #pragma unroll 4
    for (int c0 = 0; c0 < C1; c0 += 4) {
      const int ca = c0 + 2 * hf, cb = ca + 1;
      v2f a;
      a.x = pa[ca * K2 + k];                  // A[m][k*256+c] = W2[co][c][k]
      a.y = pa[cb * K2 + k];
      v2f b0 = *(const v2f*)&bsc[lo * BSTRIDE + ca];          // ds_load_b64
      v2f b1 = *(const v2f*)&bsc[(16 + lo) * BSTRIDE + ca];
      v2f b2 = *(const v2f*)&bsc[(32 + lo) * BSTRIDE + ca];
      acc0 = wmma_f32x4(a, b0, acc0);
      acc1 = wmma_f32x4(a, b1, acc1);
      acc2 = wmma_f32x4(a, b2, acc2);
    }
    __builtin_amdgcn_s_wait_tensorcnt(0);     // k+1 panel landed (wave 0)
    __syncthreads();                          // all waves see it
  }

  #pragma unroll
  for (int j = 0; j < 8; ++j) {
    const int m = m0 + wave * 16 + j + 8 * hf;   // co
    xseq[((size_t)b * L2OUT + (n0 + lo)) * C2 + m] = acc0[j];
    xseq[((size_t)b * L2OUT + (n0 + 16 + lo)) * C2 + m] = acc1[j];
    xseq[((size_t)b * L2OUT + (n0 + 32 + lo)) * C2 + m] = acc2[j];
  }
}

// ---------------------------------------------------------------------------
// K4: topo conv: pts[b][j][p] = sum_{c<512,k<30} xseq[b][6j+k][c]*wt[p][c][k] + bt[p]
// grid (20, 32), block 32, shfl-xor reduction.
// ---------------------------------------------------------------------------
__global__ void topo_conv_kernel(const float* __restrict__ xseq,
                                 const float* __restrict__ wt,
                                 const float* __restrict__ bt,
                                 float* __restrict__ pts) {
  const int j = blockIdx.x;
  const int b = blockIdx.y;
  const int lane = threadIdx.x & 31;
  for (int p = 0; p < 3; ++p) {
    float acc = 0.f;
    for (int c = lane; c < C2; c += 32) {
      const float* wp = wt + ((size_t)p * C2 + c) * 30;
      const float* xp = xseq + ((size_t)b * L2OUT + 6 * j) * C2 + c;
      #pragma unroll
      for (int k = 0; k < 30; ++k) acc += xp[(size_t)k * C2] * wp[k];
    }
    for (int off = 16; off >= 1; off >>= 1) acc += __shfl_xor(acc, off, 32);
    if (lane == 0) pts[((size_t)b * 20 + j) * 3 + p] = acc + bt[p];
  }
}

// ---------------------------------------------------------------------------
// K5: Prim's MST (20 pts in R^3) -> sorted 19 edge weights -> lin1 + relu
// One wave per batch. grid 32, block 32.
// ---------------------------------------------------------------------------
__global__ void mst_lin1_kernel(const float* __restrict__ pts,
                                const float* __restrict__ lin1,
                                float* __restrict__ t1) {
  const int b = blockIdx.x;
  const int lane = threadIdx.x & 31;
  __shared__ float sx[32], sy[32], sz[32];
  __shared__ float mw[19];

  float px = 0.f, py = 0.f, pz = 0.f;
  if (lane < 20) {
    px = pts[((size_t)b * 20 + lane) * 3 + 0];
    py = pts[((size_t)b * 20 + lane) * 3 + 1];
    pz = pts[((size_t)b * 20 + lane) * 3 + 2];
  }
  sx[lane] = px; sy[lane] = py; sz[lane] = pz;
  __syncthreads();

  float dx = px - sx[0], dy = py - sy[0], dz = pz - sz[0];
  float dist = sqrtf(dx * dx + dy * dy + dz * dz + 1e-12f);
  bool in_tree = (lane == 0);

  for (int it = 0; it < 19; ++it) {
    float v = (in_tree || lane >= 20) ? INFINITY : dist;
    int idx = lane;
    for (int off = 16; off >= 1; off >>= 1) {   // argmin, ties -> lowest index
      float ov = __shfl_xor(v, off, 32);
      int oi = __shfl_xor(idx, off, 32);
      if (ov < v || (ov == v && oi < idx)) { v = ov; idx = oi; }
    }
    if (lane == 0) mw[it] = v;
    const float qx = sx[idx], qy = sy[idx], qz = sz[idx];
    dx = px - qx; dy = py - qy; dz = pz - qz;
    dist = fminf(dist, sqrtf(dx * dx + dy * dy + dz * dz + 1e-12f));
    if (lane == idx) in_tree = true;
  }
  __syncthreads();
  if (lane == 0) {                               // insertion sort 19 values
    for (int i = 1; i < 19; ++i) {
      float key = mw[i];
      int k = i - 1;
      while (k >= 0 && mw[k] > key) { mw[k + 1] = mw[k]; --k; }
      mw[k + 1] = key;
    }
  }
  __syncthreads();
  for (int o = lane; o < 144; o += 32) {         // lin1 + relu
    float acc = 0.f;
    #pragma unroll
    for (int i = 0; i < 19; ++i) acc += mw[i] * lin1[o * 19 + i];
    t1[(size_t)b * 144 + o] = fmaxf(acc, 0.f);
  }
}

// ---------------------------------------------------------------------------
// K6: lin2 + relu as WMMA GEMM: out2[b][o] = relu(sum_i t1[b][i]*L2w[o][i])
// M=73728, N=32, K=144.  grid (2, 4608), block 32.
// ---------------------------------------------------------------------------
__global__ void lin2_wmma_kernel(const float* __restrict__ t1,
                                 const float* __restrict__ L2w,
                                 float* __restrict__ out2) {
  const int n0 = blockIdx.x * 16;
  const int m0 = blockIdx.y * 16;
  const int lane = threadIdx.x & 31;
  const int lo = lane & 15;
  const int hf = lane >> 4;

  v8f acc = {};
  #pragma unroll 4
  for (int c0 = 0; c0 < 144; c0 += 4) {
    const int ca = c0 + 2 * hf, cb = ca + 1;
    v2f a, b;
    a.x = L2w[(size_t)(m0 + lo) * 144 + ca];
    a.y = L2w[(size_t)(m0 + lo) * 144 + cb];
    b.x = t1[(size_t)(n0 + lo) * 144 + ca];
    b.y = t1[(size_t)(n0 + lo) * 144 + cb];
    acc = wmma_f32x4(a, b, acc);
  }
  #pragma unroll
  for (int j = 0; j < 8; ++j) {
    const int m = m0 + j + 8 * hf;
    out2[(size_t)(n0 + lo) * 73728 + m] = fmaxf(acc[j], 0.f);
  }
}

// ---------------------------------------------------------------------------
extern "C" void kernel_launch(void* const* d_in, const int* in_sizes, int n_in,
                              void* d_out, int out_size, void* d_ws, size_t ws_size,
                              hipStream_t stream) {
  const int* x = (const int*)d_in[0];
  const float* emb = (const float*)d_in[1];
  const float* w1 = (const float*)d_in[2];
  const float* w2 = (const float*)d_in[3];
  const float* wt = (const float*)d_in[4];
  const float* bt = (const float*)d_in[5];
  const float* l1 = (const float*)d_in[6];
  const float* l2 = (const float*)d_in[7];
  float* out = (float*)d_out;
  float* ws = (float*)d_ws;

  float* T = ws + WS_T;
  float* H1 = ws + WS_H1;
  float* pts = ws + WS_PTS;
  float* t1 = ws + WS_T1;

  build_table_kernel<<<dim3(K1, VOCAB / 16, C1 / 16), 32, 0, stream>>>(emb, w1, T);
  conv1_gather_kernel<<<dim3(L1OUT, BATCH), 64, 0, stream>>>(x, T, H1);
  conv2_wmma_kernel<<<dim3(3, 4, BATCH), 256, 0, stream>>>(H1, w2, out);
  topo_conv_kernel<<<dim3(20, BATCH), 32, 0, stream>>>(out, wt, bt, pts);
  mst_lin1_kernel<<<BATCH, 32, 0, stream>>>(pts, l1, t1);
  lin2_wmma_kernel<<<dim3(2, 73728 / 16), 32, 0, stream>>>(t1, l2, out + XSEQ_ELEMS);
}